// GATEncoder_v3_29557964931456
// MI455X (gfx1250) — compile-verified
//
#include <hip/hip_runtime.h>
#include <hip/hip_bf16.h>

// ---------------------------------------------------------------------------
// GATv2 2-layer encoder for MI455X (gfx1250, wave32).
//   - Dense node transforms:  V_WMMA_F32_16X16X4_F32, A-tile staged in LDS
//     via CDNA5 async global->LDS copies (ASYNCcnt) + ds_load for operands
//   - Edge attention: float4 gathers, ordered-uint atomic max (segment max),
//     global_atomic_add_f32 (segment sum / aggregate)
// ---------------------------------------------------------------------------

typedef float v2f __attribute__((ext_vector_type(2)));
typedef float v8f __attribute__((ext_vector_type(8)));

#define NEG_SLOPE 0.2f
#define AS3 __attribute__((address_space(3)))

// ---------------- order-preserving float <-> uint for atomic max ------------
__device__ __forceinline__ unsigned f2ord(float f) {
    unsigned u = __float_as_uint(f);
    return (u & 0x80000000u) ? ~u : (u | 0x80000000u);
}
__device__ __forceinline__ float ord2f(unsigned u) {
    unsigned v = (u & 0x80000000u) ? (u ^ 0x80000000u) : ~u;
    return __uint_as_float(v);
}

// ---------------------------------------------------------------------------
// WMMA GEMM:  Y[rows, cols] = X[rows, KDIM] @ W[KDIM, cols] + bias[cols]
// Block = 128 threads (4 waves) -> one 16-row strip, 4 16-col tiles.
// A-tile (16 x KDIM fp32) staged to LDS once per block with
// global_load_async_to_lds_b128; WMMA A operands then come from LDS.
// LDS row stride 68 floats -> the 32 lanes' b64 reads hit all 64 banks.
// A-matrix VGPR layout (ISA 7.12.2): lanes 0-15 hold (K,K+1) of row M=lane,
//   lanes 16-31 hold (K+2,K+3) of row M=lane-16. B symmetric; C/D: VGPR r is
//   row M = r + 8*(lane>=16), col N = lane&15.
// ---------------------------------------------------------------------------
template <int KDIM>
__global__ __launch_bounds__(128) void gemm16_wmma_f32(
    const float* __restrict__ X, const float* __restrict__ W,
    const float* __restrict__ bias, float* __restrict__ Y, int rows,
    int cols) {
    constexpr int LDA = 68;                 // padded row stride (floats)
    __shared__ float As[16 * LDA];

    const int ngroups = cols >> 6;          // 4 tiles (64 cols) per block
    const int bm = blockIdx.x / ngroups;
    const int bg = blockIdx.x - bm * ngroups;
    const int m0 = bm << 4;
    const int tid = threadIdx.x;
    const int wave = tid >> 5;
    const int lane = tid & 31;

    // ---- stage A tile: 16 rows x KDIM floats, async global -> LDS --------
    {
        const unsigned as_base =
            (unsigned)(unsigned long long)(AS3 float*)&As[0];
        constexpr int NF4 = 16 * KDIM / 4;  // float4 chunks in the tile
#pragma unroll
        for (int i = tid; i < NF4; i += 128) {
            const int row = i / (KDIM / 4);
            const int c4 = i - row * (KDIM / 4);
            int gr = m0 + row;
            if (gr >= rows) gr = rows - 1;
            const unsigned goff =
                (unsigned)(((size_t)gr * KDIM + c4 * 4) * sizeof(float));
            const unsigned loff =
                as_base + (unsigned)((row * LDA + c4 * 4) * sizeof(float));
            asm volatile("global_load_async_to_lds_b128 %0, %1, %2"
                         :
                         : "v"(loff), "v"(goff), "s"(X)
                         : "memory");
        }
        asm volatile("s_wait_asynccnt 0x0" ::: "memory");
        __syncthreads();
    }

    const int n0 = (bg * 4 + wave) << 4;
    const int half = lane >> 4;             // 0: lanes 0-15, 1: lanes 16-31
    const int l15 = lane & 15;
    const int colB = n0 + l15;
    const float* __restrict__ arow = &As[l15 * LDA + (half << 1)];

    v8f acc = {};
#pragma unroll
    for (int k = 0; k < KDIM; k += 4) {
        v2f a;                              // ds_load_b64 from LDS
        a.x = arow[k];
        a.y = arow[k + 1];
        v2f b;
        const int kb = k + (half << 1);
        b.x = W[(size_t)kb * cols + colB];
        b.y = W[(size_t)(kb + 1) * cols + colB];
        acc = __builtin_amdgcn_wmma_f32_16x16x4_f32(
            /*neg_a=*/false, a, /*neg_b=*/false, b,
            /*c_mod=*/(short)0, acc, /*reuse_a=*/false, /*reuse_b=*/false);
    }

    const float bv = bias[colB];
#pragma unroll
    for (int r = 0; r < 8; ++r) {
        const int row = m0 + r + (half << 3);
        if (row < rows) Y[(size_t)row * cols + colB] = acc[r] + bv;
    }
}

// ---------------------------------------------------------------------------
// Edge pass 1: score[e,h] = att[h,:] . leaky_relu(xl[src,h,:] + xr[dst,h,:])
//              segment-max into m_ord[dst,h]  (ordered-uint atomic max)
// One thread per (edge, head); heads adjacent -> coalesced 64B/256B chunks.
// ---------------------------------------------------------------------------
template <int C>
__global__ void edge_score_max(const long long* __restrict__ ei, int E, int n,
                               const float* __restrict__ xl,
                               const float* __restrict__ xr,
                               const float* __restrict__ att,
                               unsigned* __restrict__ m_ord,
                               float* __restrict__ score) {
    constexpr int H = 4;
    const long long total = ((long long)E + n) * H;
    const long long t = (long long)blockIdx.x * blockDim.x + threadIdx.x;
    if (t >= total) return;
    const int h = (int)(t & (H - 1));
    const long long e = t >> 2;
    long long s, d;
    if (e < E) { s = ei[e]; d = ei[E + e]; } else { s = d = e - E; }

    const float4* __restrict__ pl = (const float4*)(xl + ((size_t)s * H + h) * C);
    const float4* __restrict__ pr = (const float4*)(xr + ((size_t)d * H + h) * C);
    const float4* __restrict__ pa = (const float4*)(att + (size_t)h * C);

    float sc = 0.f;
#pragma unroll
    for (int c4 = 0; c4 < C / 4; ++c4) {
        const float4 a = pl[c4];
        const float4 b = pr[c4];
        const float4 w = pa[c4];
        float v;
        v = a.x + b.x; v = v > 0.f ? v : NEG_SLOPE * v; sc += w.x * v;
        v = a.y + b.y; v = v > 0.f ? v : NEG_SLOPE * v; sc += w.y * v;
        v = a.z + b.z; v = v > 0.f ? v : NEG_SLOPE * v; sc += w.z * v;
        v = a.w + b.w; v = v > 0.f ? v : NEG_SLOPE * v; sc += w.w * v;
    }
    score[t] = sc;
    atomicMax(&m_ord[(size_t)d * H + h], f2ord(sc));
}

// ---------------------------------------------------------------------------
// Edge pass 2: ex = exp(score - m[dst]); score <- ex; denom[dst,h] += ex
// ---------------------------------------------------------------------------
__global__ void edge_exp_sum(const long long* __restrict__ ei, int E, int n,
                             const unsigned* __restrict__ m_ord,
                             float* __restrict__ score,
                             float* __restrict__ denom) {
    constexpr int H = 4;
    const long long total = ((long long)E + n) * H;
    const long long t = (long long)blockIdx.x * blockDim.x + threadIdx.x;
    if (t >= total) return;
    const int h = (int)(t & (H - 1));
    const long long e = t >> 2;
    long long d;
    if (e < E) { d = ei[E + e]; } else { d = e - E; }

    const float m = ord2f(m_ord[(size_t)d * H + h]);
    const float ex = __expf(score[t] - m);
    score[t] = ex;
    atomicAdd(&denom[(size_t)d * H + h], ex);
}

// ---------------------------------------------------------------------------
// Edge pass 3: out[dst] += head_scale * (ex / (denom[dst]+eps)) * xl[src,h,:]
// concat=true : out is [n, H*C], head h writes slice h*C..h*C+C
// concat=false: out is [n, C],   all heads accumulate (head_scale = 1/H)
// ---------------------------------------------------------------------------
template <int C, bool CONCAT>
__global__ void edge_aggregate(const long long* __restrict__ ei, int E, int n,
                               const float* __restrict__ ex,
                               const float* __restrict__ denom,
                               const float* __restrict__ xl,
                               float* __restrict__ out, float head_scale) {
    constexpr int H = 4;
    const long long total = ((long long)E + n) * H;
    const long long t = (long long)blockIdx.x * blockDim.x + threadIdx.x;
    if (t >= total) return;
    const int h = (int)(t & (H - 1));
    const long long e = t >> 2;
    long long s, d;
    if (e < E) { s = ei[e]; d = ei[E + e]; } else { s = d = e - E; }

    const float alpha =
        ex[t] / (denom[(size_t)d * H + h] + 1e-16f) * head_scale;
    const float4* __restrict__ pl = (const float4*)(xl + ((size_t)s * H + h) * C);
    float* __restrict__ po =
        CONCAT ? (out + ((size_t)d * H + h) * C) : (out + (size_t)d * C);

#pragma unroll
    for (int c4 = 0; c4 < C / 4; ++c4) {
        const float4 a = pl[c4];
        atomicAdd(po + 4 * c4 + 0, alpha * a.x);
        atomicAdd(po + 4 * c4 + 1, alpha * a.y);
        atomicAdd(po + 4 * c4 + 2, alpha * a.z);
        atomicAdd(po + 4 * c4 + 3, alpha * a.w);
    }
}

// ---------------------------------------------------------------------------
// out[i] = relu(in[i] + bias[i % COLS])   (in-place safe; COLS power of two)
// ---------------------------------------------------------------------------
template <int COLS>
__global__ void bias_relu(const float* __restrict__ in,
                          const float* __restrict__ bias,
                          float* __restrict__ out, long long total) {
    const long long t = (long long)blockIdx.x * blockDim.x + threadIdx.x;
    if (t >= total) return;
    const int c = (int)(t & (COLS - 1));
    const float v = in[t] + bias[c];
    out[t] = v > 0.f ? v : 0.f;
}

// ---------------------------------------------------------------------------
extern "C" void kernel_launch(void* const* d_in, const int* in_sizes, int n_in,
                              void* d_out, int out_size, void* d_ws,
                              size_t ws_size, hipStream_t stream) {
    const float*     x  = (const float*)d_in[0];
    const long long* ei = (const long long*)d_in[1];
    // d_in[2] frame_mask: unused by the reference
    const float* Wl1  = (const float*)d_in[3];
    const float* bl1  = (const float*)d_in[4];
    const float* Wr1  = (const float*)d_in[5];
    const float* br1  = (const float*)d_in[6];
    const float* att1 = (const float*)d_in[7];
    const float* bo1  = (const float*)d_in[8];
    const float* Wl2  = (const float*)d_in[9];
    const float* bl2  = (const float*)d_in[10];
    const float* Wr2  = (const float*)d_in[11];
    const float* br2  = (const float*)d_in[12];
    const float* att2 = (const float*)d_in[13];
    const float* bo2  = (const float*)d_in[14];

    const int NIN = 64, H = 4, C1 = 16, HC1 = 64, NOUT = 64, HNOUT = 256;
    const int n = in_sizes[0] / NIN;
    const int E = in_sizes[1] / 2;
    const long long EE = (long long)E + n;

    // ---- workspace carve-out --------------------------------------------
    char* wsp = (char*)d_ws;
    auto carve = [&](size_t bytes) -> char* {
        char* p = wsp;
        wsp += (bytes + 255) & ~(size_t)255;
        return p;
    };
    float*    xl1   = (float*)carve((size_t)n * HC1 * 4);
    float*    xr1   = (float*)carve((size_t)n * HC1 * 4);
    float*    hbuf  = (float*)carve((size_t)n * HC1 * 4);   // layer-1 agg/relu
    float*    xl2   = (float*)carve((size_t)n * HNOUT * 4);
    float*    xr2   = (float*)carve((size_t)n * HNOUT * 4);
    float*    score = (float*)carve((size_t)EE * H * 4);
    unsigned* m_ord = (unsigned*)carve((size_t)n * H * 4);
    float*    denom = (float*)carve((size_t)n * H * 4);
    float*    out   = (float*)d_out;

    const int BT = 256;
    auto gemm = [&](const float* X, const float* W, const float* b, float* Y,
                    int rows, int cols) {   // KDIM fixed at 64 for both layers
        const int mtiles = (rows + 15) / 16;
        const int blocks = mtiles * (cols / 64);
        gemm16_wmma_f32<64><<<blocks, 128, 0, stream>>>(X, W, b, Y, rows, cols);
    };
    const long long eThreads = EE * H;
    const int eBlocks = (int)((eThreads + BT - 1) / BT);

    // ======================= Layer 1 (64 -> 4x16, concat) ==================
    gemm(x, Wl1, bl1, xl1, n, HC1);
    gemm(x, Wr1, br1, xr1, n, HC1);

    hipMemsetAsync(m_ord, 0, (size_t)n * H * 4, stream);
    hipMemsetAsync(denom, 0, (size_t)n * H * 4, stream);
    hipMemsetAsync(hbuf, 0, (size_t)n * HC1 * 4, stream);

    edge_score_max<C1><<<eBlocks, BT, 0, stream>>>(ei, E, n, xl1, xr1, att1,
                                                   m_ord, score);
    edge_exp_sum<<<eBlocks, BT, 0, stream>>>(ei, E, n, m_ord, score, denom);
    edge_aggregate<C1, true><<<eBlocks, BT, 0, stream>>>(ei, E, n, score, denom,
                                                         xl1, hbuf, 1.0f);
    {
        const long long tot = (long long)n * HC1;
        bias_relu<HC1><<<(int)((tot + BT - 1) / BT), BT, 0, stream>>>(
            hbuf, bo1, hbuf, tot);
    }

    // ======================= Layer 2 (64 -> 4x64, mean) ====================
    gemm(hbuf, Wl2, bl2, xl2, n, HNOUT);
    gemm(hbuf, Wr2, br2, xr2, n, HNOUT);

    hipMemsetAsync(m_ord, 0, (size_t)n * H * 4, stream);
    hipMemsetAsync(denom, 0, (size_t)n * H * 4, stream);
    hipMemsetAsync(out, 0, (size_t)out_size * 4, stream);

    edge_score_max<NOUT><<<eBlocks, BT, 0, stream>>>(ei, E, n, xl2, xr2, att2,
                                                     m_ord, score);
    edge_exp_sum<<<eBlocks, BT, 0, stream>>>(ei, E, n, m_ord, score, denom);
    edge_aggregate<NOUT, false><<<eBlocks, BT, 0, stream>>>(
        ei, E, n, score, denom, xl2, out, 1.0f / (float)H);
    {
        const long long tot = (long long)n * NOUT;
        bias_relu<NOUT><<<(int)((tot + BT - 1) / BT), BT, 0, stream>>>(
            out, bo2, out, tot);
    }
}